// TPSGridGen_69337952027246
// MI455X (gfx1250) — compile-verified
//
#include <hip/hip_runtime.h>
#include <hip/hip_bf16.h>
#include <math.h>

// Problem constants (match reference)
#define HH   512
#define WW   512
#define KK   16          // control points
#define KP3  19          // K+3
#define KPAD 20          // K+3 padded to multiple of 4 (WMMA K-chunks of 4)
#define BB   64
#define NCOL 128         // B*2 columns, col c = 2*b + d
#define HWN  (HH*WW)

#define BLDS 136         // padded LDS row stride for B-matrix (conflict-free)
#define SROW 68          // padded LDS row stride for staging half-tile (64 cols + 4)

typedef __attribute__((ext_vector_type(2))) float v2f;
typedef __attribute__((ext_vector_type(8))) float v8f;

__device__ __forceinline__ float cpx_of(int k) { return -1.0f + (2.0f/3.0f)*(float)(k >> 2); }
__device__ __forceinline__ float cpy_of(int k) { return -1.0f + (2.0f/3.0f)*(float)(k & 3); }
__device__ __forceinline__ float tps_u(float r2) { return r2 * logf(r2 + 1e-6f); }

// ---------------------------------------------------------------------------
// Setup: build delta(19x19) from the fixed control lattice, invert it
// (Gauss-Jordan w/ partial pivoting on thread 0 — trivial cost), then compute
// coeff[b,k,d] = sum_j invDelta[k,j] * src[b,j,d] and lay it out as the WMMA
// B-matrix: bmat[k][c] with c = 2*b + d, K padded to 20 with a zero row.
// ---------------------------------------------------------------------------
__global__ void tps_setup_kernel(const float* __restrict__ sp, float* __restrict__ bmat) {
    __shared__ float aug[KP3][2*KP3];
    __shared__ float invD[KP3][KP3];
    const int tid = threadIdx.x;

    if (tid == 0) {
        for (int i = 0; i < KP3; ++i) {
            for (int j = 0; j < KP3; ++j) {
                float v;
                if (i < KK && j < KK) {
                    float dx = cpx_of(i) - cpx_of(j);
                    float dy = cpy_of(i) - cpy_of(j);
                    v = tps_u(dx*dx + dy*dy);          // U(0)=0 on diagonal
                } else if (i < KK) {
                    v = (j == KK) ? 1.0f : ((j == KK+1) ? cpx_of(i) : cpy_of(i));
                } else if (j < KK) {
                    v = (i == KK) ? 1.0f : ((i == KK+1) ? cpx_of(j) : cpy_of(j));
                } else {
                    v = 0.0f;                          // bottom-right 3x3 zero
                }
                aug[i][j] = v;
                aug[i][KP3 + j] = (i == j) ? 1.0f : 0.0f;
            }
        }
        // Gauss-Jordan with partial pivoting
        for (int p = 0; p < KP3; ++p) {
            int piv = p; float best = fabsf(aug[p][p]);
            for (int r = p + 1; r < KP3; ++r) {
                float a = fabsf(aug[r][p]);
                if (a > best) { best = a; piv = r; }
            }
            if (piv != p) {
                for (int j = 0; j < 2*KP3; ++j) {
                    float t = aug[p][j]; aug[p][j] = aug[piv][j]; aug[piv][j] = t;
                }
            }
            float inv = 1.0f / aug[p][p];
            for (int j = 0; j < 2*KP3; ++j) aug[p][j] *= inv;
            for (int r = 0; r < KP3; ++r) {
                if (r == p) continue;
                float f = aug[r][p];
                for (int j = 0; j < 2*KP3; ++j) aug[r][j] -= f * aug[p][j];
            }
        }
        for (int i = 0; i < KP3; ++i)
            for (int j = 0; j < KP3; ++j)
                invD[i][j] = aug[i][KP3 + j];
    }
    __syncthreads();

    const int c = tid;                 // 0..127 : column = (b,d)
    if (c < NCOL) {
        const int b = c >> 1, d = c & 1;
        float spv[KK];
        #pragma unroll
        for (int j = 0; j < KK; ++j) spv[j] = sp[b*(KK*2) + j*2 + d];
        for (int i = 0; i < KP3; ++i) {
            float acc = 0.0f;
            #pragma unroll
            for (int j = 0; j < KK; ++j) acc += invD[i][j] * spv[j];   // rows >=16 of Pm are zero
            bmat[i*NCOL + c] = acc;
        }
        bmat[KP3*NCOL + c] = 0.0f;     // zero pad row (K=19 slot)
    }
}

// ---------------------------------------------------------------------------
// Main: GEMM [HW x 20] x [20 x 128] via V_WMMA_F32_16X16X4_F32.
// Each 256-thread block = 8 waves; each wave owns a 16-row n-tile and loops
// over all 8 column tiles (5 WMMA K-chunks each). Results are transposed
// through per-wave LDS staging so global stores are fully coalesced float2s.
// ---------------------------------------------------------------------------
__global__ void __launch_bounds__(256)
tps_main_kernel(const float* __restrict__ bmat, float* __restrict__ out) {
    __shared__ float Bls[KPAD * BLDS];            // 20 x 136  (~10.6 KB)
    __shared__ float stage[8][16 * SROW];         // 8 waves x 16 x 68 (~34.8 KB)

    const int tid = threadIdx.x;

    // Cooperative load of the 20x128 B-matrix into padded LDS rows.
    for (int idx = tid; idx < KPAD * NCOL; idx += 256) {
        int r = idx >> 7, cc = idx & 127;
        Bls[r * BLDS + cc] = bmat[idx];
    }
    __syncthreads();

    const int lane = tid & 31;
    const int wv   = tid >> 5;
    const int m    = lane & 15;
    const bool hi  = lane >= 16;
    const int n0   = blockIdx.x * 128 + wv * 16;
    const int n    = n0 + m;

    const float px = -1.0f + 2.0f * (float)(n & (WW - 1)) * (1.0f / (float)(WW - 1));
    const float py = -1.0f + 2.0f * (float)(n >> 9)       * (1.0f / (float)(HH - 1));

    // A-fragments, ISA f32 16x4 layout: lanes 0-15 hold K=4kb+{0,1} in {v.x,v.y},
    // lanes 16-31 hold K=4kb+{2,3}. Chunk 4 carries the affine rows [1, x, y, 0].
    v2f A[5];
    #pragma unroll
    for (int kb = 0; kb < 4; ++kb) {
        int k0 = kb * 4 + (hi ? 2 : 0);
        float dx0 = px - cpx_of(k0),     dy0 = py - cpy_of(k0);
        float dx1 = px - cpx_of(k0 + 1), dy1 = py - cpy_of(k0 + 1);
        float r20 = dx0*dx0 + dy0*dy0;
        float r21 = dx1*dx1 + dy1*dy1;
        A[kb].x = r20 * logf(r20 + 1e-6f);
        A[kb].y = r21 * logf(r21 + 1e-6f);
    }
    A[4].x = hi ? py   : 1.0f;    // rows K=16 (ones) / K=18 (y)
    A[4].y = hi ? 0.0f : px;      // rows K=17 (x)    / K=19 (zero pad)

    float* st = &stage[wv][0];
    const int colLane = lane & 15;
    const int rbase   = hi ? 8 : 0;

    // Two halves of 4 column-tiles each (keeps staging LDS small).
    for (int h2 = 0; h2 < 2; ++h2) {
        #pragma unroll
        for (int t = 0; t < 4; ++t) {
            const int c0  = (h2 * 4 + t) * 16;
            const int col = c0 + colLane;
            v8f C = {0.f, 0.f, 0.f, 0.f, 0.f, 0.f, 0.f, 0.f};
            #pragma unroll
            for (int kb = 0; kb < 5; ++kb) {
                int r0 = kb * 4 + (hi ? 2 : 0);
                v2f Bv;
                Bv.x = Bls[r0 * BLDS + col];
                Bv.y = Bls[(r0 + 1) * BLDS + col];
                C = __builtin_amdgcn_wmma_f32_16x16x4_f32(
                        /*neg_a=*/false, A[kb], /*neg_b=*/false, Bv,
                        /*c_mod=*/(short)0, C, /*reuse_a=*/false, /*reuse_b=*/false);
            }
            // C layout: lane holds column `col`, rows rbase+v in C[v].
            const int lc = t * 16 + colLane;
            #pragma unroll
            for (int v = 0; v < 8; ++v)
                st[(rbase + v) * SROW + lc] = C[v];
        }
        __syncthreads();   // staging writes visible (cross-lane via LDS)

        // Coalesced writeback: lanes 0-15 -> batch b0, lanes 16-31 -> b0+1;
        // each lane stores float2 (d=0,1) for its n -> two 128B segments/store.
        #pragma unroll
        for (int bp = 0; bp < 16; ++bp) {
            const int b  = h2 * 32 + 2 * bp + (hi ? 1 : 0);
            const int lc = 4 * bp + (hi ? 2 : 0);                // 2*b - h2*64
            v2f val = *(const v2f*)&st[m * SROW + lc];
            float* gp = out + (size_t)b * (size_t)(HWN * 2) + (size_t)(n0 + m) * 2;
            *(v2f*)gp = val;
        }
        __syncthreads();   // protect next half's staging writes
    }
}

extern "C" void kernel_launch(void* const* d_in, const int* in_sizes, int n_in,
                              void* d_out, int out_size, void* d_ws, size_t ws_size,
                              hipStream_t stream) {
    (void)in_sizes; (void)n_in; (void)out_size; (void)ws_size;
    const float* src = (const float*)d_in[0];   // [B, K, 2] fp32
    float* out       = (float*)d_out;           // [B, H, W, 2] fp32
    float* bmat      = (float*)d_ws;            // [20][128] fp32 coeff B-matrix

    tps_setup_kernel<<<1, 128, 0, stream>>>(src, bmat);

    const int nBlocks = HWN / 128;              // 2048 blocks, 8 waves each
    tps_main_kernel<<<nBlocks, 256, 0, stream>>>(bmat, out);
}